// SAGEMeanConv_89876485636135
// MI455X (gfx1250) — compile-verified
//
#include <hip/hip_runtime.h>
#include <hip/hip_bf16.h>

// GraphSAGE mean-aggregation conv for MI455X (gfx1250, wave32).
//
//   h_self  = feat @ W                  (fp32 WMMA 16x16x4 GEMM, B staged in LDS async)
//   out     = h_self                    (seeded by GEMM kernel)
//   out[d] += h_self[src[e]]  for all e (wave-per-edge scatter, f32 atomics, L2-resident)
//   out     = relu(out / (deg+1))       (vectorized finalize)

#define N_NODES   100000
#define IN_FEATS  256
#define OUT_FEATS 128
#define LDS_K     260   // padded K-stride: ds_load_b64 pattern hits all 64 banks once

typedef __attribute__((ext_vector_type(2))) float v2f;
typedef __attribute__((ext_vector_type(8))) float v8f;

typedef __attribute__((address_space(1))) int as1_i32;
typedef __attribute__((address_space(3))) int as3_i32;

// ---- CDNA5 async global->LDS copy (ASYNCcnt path), with safe fallbacks ----
__device__ __forceinline__ void async_copy_f32(const float* g, float* l) {
#if __has_builtin(__builtin_amdgcn_global_load_async_to_lds_b32)
    // Probe-learned signature: (AS1 int*, AS3 int*, imm offset, imm cpol)
    __builtin_amdgcn_global_load_async_to_lds_b32(
        (as1_i32*)const_cast<float*>(g),
        (as3_i32*)l, 0, 0);
#else
    *l = *g;                       // sync fallback
#endif
}

__device__ __forceinline__ void async_wait0() {
#if __has_builtin(__builtin_amdgcn_s_wait_asynccnt)
    __builtin_amdgcn_s_wait_asynccnt(0);
#else
    asm volatile("s_wait_asynccnt 0" ::: "memory");
#endif
}

// ---------------------------------------------------------------------------
// Kernel 1: h_self = feat @ W with V_WMMA_F32_16X16X4_F32.
// Block = 256 threads = 8 waves = a 128-row x 16-col output panel; all 8 waves
// share one 256x16 slice of W, staged once into LDS (transposed + padded) by
// GLOBAL_LOAD_ASYNC_TO_LDS_B32 and read back as bank-conflict-free ds_load_b64.
//
// Fragment layouts (wave32):
//  A (16x4 f32):  lane L<16: M=L, K=k..k+1;  lane L+16: M=L, K=k+2..k+3
//  B (4x16 f32):  lane L<16: N=L, K=k..k+1;  lane L+16: N=L, K=k+2..k+3
//  C/D (16x16):   VGPR r: lanes 0-15 -> M=r, lanes 16-31 -> M=r+8; N = lane%16
// ---------------------------------------------------------------------------
__global__ __launch_bounds__(256) void sage_gemm_wmma(
    const float* __restrict__ feat,   // [N_NODES, IN_FEATS] row-major
    const float* __restrict__ W,      // [IN_FEATS, OUT_FEATS] row-major
    float* __restrict__ h_self,       // [N_NODES, OUT_FEATS]
    float* __restrict__ out)          // [N_NODES, OUT_FEATS] (accumulator seed)
{
    __shared__ float ldsB[16 * LDS_K];          // ldsB[n*LDS_K + k] = W[k][col0+n]

    const int wave = threadIdx.x >> 5;
    const int lane = threadIdx.x & 31;
    const int colt = blockIdx.x & 7;            // column tile 0..7
    const int rowb = blockIdx.x >> 3;           // 128-row panel 0..781
    const int col0 = colt * 16;
    const int row0 = rowb * 128 + wave * 16;

    // ---- stage W[:, col0:col0+16] (16 KB) into LDS, transposed, async ----
    // 4096 floats / 256 threads = 16 b32 copies per thread.
    {
        const int t = threadIdx.x;
#pragma unroll
        for (int i = 0; i < 16; ++i) {
            const int l = i * 256 + t;          // 0..4095
            const int k = l >> 4;
            const int n = l & 15;
            async_copy_f32(W + (size_t)k * OUT_FEATS + col0 + n,
                           &ldsB[n * LDS_K + k]);
        }
        async_wait0();
    }
    __syncthreads();

    if (row0 < N_NODES) {                       // tail panel: 16 | N_NODES, whole-tile guard
        const int mn   = lane & 15;             // M index (A) / N index (B)
        const int koff = (lane >> 4) * 2;       // 0 (lanes 0-15) or 2 (lanes 16-31)

        const float* __restrict__ Arow = feat + (size_t)(row0 + mn) * IN_FEATS + koff;
        const float* __restrict__ Brow = &ldsB[mn * LDS_K + koff];

        v8f acc = {};
#pragma unroll 8
        for (int k = 0; k < IN_FEATS; k += 4) {
            v2f a = *(const v2f*)(Arow + k);    // global_load_b64
            v2f b = *(const v2f*)(Brow + k);    // ds_load_b64, conflict-free (pad 260)
            acc = __builtin_amdgcn_wmma_f32_16x16x4_f32(
                /*neg_a=*/false, a, /*neg_b=*/false, b,
                /*c_mod=*/(short)0, acc, /*reuse_a=*/false, /*reuse_b=*/false);
        }

        // VGPR r -> row (lane<16 ? r : r+8), col = lane%16 (coalesced across lanes)
        const int mbase = (lane >> 4) * 8;
#pragma unroll
        for (int r = 0; r < 8; ++r) {
            const size_t idx = (size_t)(row0 + mbase + r) * OUT_FEATS + col0 + mn;
            h_self[idx] = acc[r];
            out[idx]    = acc[r];               // seed accumulator: out starts as h_self
        }
    }
}

// ---------------------------------------------------------------------------
// Kernel 2: edge scatter. One wave per edge: 32 lanes x float4 = full 128-float
// row. Gather is a coalesced 512B global_load_b128 burst; scatter is 4
// global_atomic_add_f32 per lane. h_self and out are both L2-resident (51 MB
// each vs 192 MB L2), so this runs at L2 atomic throughput, not HBM.
// Edge indices are wave-uniform -> scalar loads; streamed edge lists get
// explicit global_prefetch hints.
// ---------------------------------------------------------------------------
__global__ __launch_bounds__(256) void sage_scatter(
    const int* __restrict__ src,
    const int* __restrict__ dst,
    const float* __restrict__ h_self,
    float* __restrict__ out,
    float* __restrict__ deg,
    int n_edges)
{
    const int wv = __builtin_amdgcn_readfirstlane(threadIdx.x >> 5);
    const int e  = blockIdx.x * 8 + wv;         // wave-uniform edge id
    if (e >= n_edges) return;
    const int lane = threadIdx.x & 31;

    if (lane == 0) {                            // prefetch edge-list stream ahead
        const int pe = (e + 4096 < n_edges) ? e + 4096 : n_edges - 1;
        __builtin_prefetch(src + pe, 0, 1);
        __builtin_prefetch(dst + pe, 0, 1);
    }

    const int s = src[e];                       // uniform -> s_load
    const int d = dst[e];

    const float4 v = *(const float4*)(h_self + (size_t)s * OUT_FEATS + lane * 4);
    float* dp = out + (size_t)d * OUT_FEATS + lane * 4;

    unsafeAtomicAdd(dp + 0, v.x);
    unsafeAtomicAdd(dp + 1, v.y);
    unsafeAtomicAdd(dp + 2, v.z);
    unsafeAtomicAdd(dp + 3, v.w);

    if (lane == 0) unsafeAtomicAdd(deg + d, 1.0f);   // in-degree count
}

// ---------------------------------------------------------------------------
// Kernel 3: out = relu(out / (deg + 1)), float4-vectorized in place.
// ---------------------------------------------------------------------------
__global__ __launch_bounds__(256) void sage_finalize(
    float* __restrict__ out,
    const float* __restrict__ deg)
{
    const size_t base = ((size_t)blockIdx.x * blockDim.x + threadIdx.x) * 4;
    if (base >= (size_t)N_NODES * OUT_FEATS) return;
    const int node = (int)(base >> 7);               // /128
    const float inv = 1.0f / (deg[node] + 1.0f);

    float4 v = *(float4*)(out + base);
    v.x = fmaxf(v.x * inv, 0.0f);
    v.y = fmaxf(v.y * inv, 0.0f);
    v.z = fmaxf(v.z * inv, 0.0f);
    v.w = fmaxf(v.w * inv, 0.0f);
    *(float4*)(out + base) = v;
}

// ---------------------------------------------------------------------------
extern "C" void kernel_launch(void* const* d_in, const int* in_sizes, int n_in,
                              void* d_out, int out_size, void* d_ws, size_t ws_size,
                              hipStream_t stream)
{
    const float* feat = (const float*)d_in[0];
    const float* W    = (const float*)d_in[1];
    const int*   src  = (const int*)d_in[2];
    const int*   dst  = (const int*)d_in[3];
    float*       out  = (float*)d_out;

    const int n_edges = in_sizes[2];

    // Workspace layout: h_self [N*128] floats, then deg [N] floats.
    float* h_self = (float*)d_ws;
    float* deg    = h_self + (size_t)N_NODES * OUT_FEATS;

    // deg must be zero every call (graph-capture-safe async memset).
    (void)hipMemsetAsync(deg, 0, (size_t)N_NODES * sizeof(float), stream);

    // 1) GEMM: 782 row-panels (128 rows) x 8 col tiles; per-wave tile guard handles tail.
    const int row_panels = (N_NODES + 127) / 128;    // 782
    sage_gemm_wmma<<<row_panels * 8, 256, 0, stream>>>(feat, W, h_self, out);

    // 2) Edge scatter: 8 edges per 256-thread block.
    sage_scatter<<<(n_edges + 7) / 8, 256, 0, stream>>>(src, dst, h_self, out, deg, n_edges);

    // 3) Finalize: N*128 floats / 4 per thread / 256 per block = 12500 blocks.
    sage_finalize<<<(N_NODES * OUT_FEATS) / 4 / 256, 256, 0, stream>>>(out, deg);
}